// DecoderCell_5248450036169
// MI455X (gfx1250) — compile-verified
//
#include <hip/hip_runtime.h>
#include <float.h>
#include <math.h>

// MI455X / gfx1250, wave32. One 128-thread workgroup (4 waves) per batch element.
// Phase 0: stage Wq_step + Wout into LDS via GLOBAL_LOAD_ASYNC_TO_LDS_B128 (ASYNCcnt).
// Phase 1: K1/V/K2 = node_emb @ W via V_WMMA_F32_16X16X4_F32 directly into LDS.
// Phase 2: 202-step serial decode loop entirely out of LDS (no per-step HBM/L2 reads
//          except the 512B prev-node embedding row).
// LDS: 66K (Wq_step) + 64K (Wout) + 3*52.1K (K1/V/K2, stride-129 padded) + ~7K misc
//    = ~294.6 KB < 320 KB. We were already at 1 WG/WGP (163KB > 160KB), so this
//    costs no occupancy and kills ~27 GB of per-step weight traffic.

typedef __attribute__((ext_vector_type(2))) float v2f;
typedef __attribute__((ext_vector_type(8))) float v8f;
typedef __attribute__((ext_vector_type(4))) int   v4i;

#define NB     1024
#define NCUST  100
#define NN     101      // depot + customers
#define EE     128
#define HH     8
#define DHD    16
#define TT     202      // 2*N steps
#define LSTR   129      // padded LDS row stride (129 mod 64 == 1 -> conflict-free)

#define NWQ    (129 * EE)   // Wq_step elements (16512)
#define NWO    (EE * EE)    // Wout elements    (16384)

#define HAVE_ASYNC_LDS \
    (__has_builtin(__builtin_amdgcn_global_load_async_to_lds_b128) && \
     __has_builtin(__builtin_amdgcn_s_wait_asynccnt))

__global__ __launch_bounds__(128) void vrp_decoder_kernel(
    const float* __restrict__ depot_xy,     // (B,2)
    const float* __restrict__ customer_xy,  // (B,100,2)
    const float* __restrict__ demand,       // (B,100)
    const float* __restrict__ node_emb,     // (B,101,128)
    const float* __restrict__ graph_emb,    // (B,128)
    const float* __restrict__ Wk1,          // (128,128)
    const float* __restrict__ Wv,
    const float* __restrict__ Wk2,
    const float* __restrict__ Wq_fixed,
    const float* __restrict__ Wout,
    const float* __restrict__ Wq_step,      // (129,128)
    float* __restrict__ out)                // cost[B] ++ ll[B]
{
    extern __shared__ float smem[];
    // 16B-aligned big arrays first (B128 transfers), then padded K/V arrays.
    float* sWq   = smem;                    // 129*128
    float* sWo   = sWq + NWQ;               // 128*128
    float* sK1   = sWo + NWO;               // 101*129
    float* sV    = sK1 + NN * LSTR;
    float* sK2   = sV  + NN * LSTR;
    float* qfix  = sK2 + NN * LSTR;         // 128
    float* prevE = qfix + EE;               // 128
    float* q1    = prevE + EE;              // 128
    float* glim  = q1 + EE;                 // 128
    float* q2v   = glim + EE;               // 128
    float* attn  = q2v + EE;                // 8*101
    float* demS  = attn + HH * NN;          // 100
    float* redf  = demS + NCUST;            // 8
    int*   redi  = (int*)(redf + 8);        // 8
    int*   vis   = redi + 8;                // 100
    int*   sti   = vis + NCUST;             // [0]=prevNode [1]=visitedCount
    float* stf   = (float*)(sti + 2);       // [0]=D [1]=cost [2]=ll [3]=cx [4]=cy [5]=depx [6]=depy

    const int b    = blockIdx.x;
    const int t    = threadIdx.x;
    const int lane = t & 31;
    const int wave = t >> 5;
    const float* NEb = node_emb + (size_t)b * NN * EE;

    // ---------------- stage step weights into LDS ----------------
#if HAVE_ASYNC_LDS
    {
        typedef __attribute__((address_space(1))) v4i gv4i_t;   // global int4
        typedef __attribute__((address_space(3))) v4i lv4i_t;   // LDS int4
        for (int i = t * 4; i < NWQ; i += 128 * 4)
            __builtin_amdgcn_global_load_async_to_lds_b128(
                (gv4i_t*)(Wq_step + i), (lv4i_t*)(sWq + i), 0, 0);
        for (int i = t * 4; i < NWO; i += 128 * 4)
            __builtin_amdgcn_global_load_async_to_lds_b128(
                (gv4i_t*)(Wout + i), (lv4i_t*)(sWo + i), 0, 0);
    }
#else
    for (int i = t * 4; i < NWQ; i += 128 * 4)
        *(float4*)(sWq + i) = *(const float4*)(Wq_step + i);
    for (int i = t * 4; i < NWO; i += 128 * 4)
        *(float4*)(sWo + i) = *(const float4*)(Wout + i);
#endif

    // ---------------- init ----------------
    prevE[t] = graph_emb[(size_t)b * EE + t];           // stage graph embedding
    if (t < NCUST) { demS[t] = demand[(size_t)b * NCUST + t]; vis[t] = 0; }
    if (t == 0) {
        sti[0] = 0; sti[1] = 0;
        stf[0] = 1.0f; stf[1] = 0.0f; stf[2] = 0.0f;
        float dx = depot_xy[(size_t)b * 2 + 0];
        float dy = depot_xy[(size_t)b * 2 + 1];
        stf[3] = dx; stf[4] = dy; stf[5] = dx; stf[6] = dy;
    }
    __syncthreads();

    // Q_fixed GEMV: qfix[f] = sum_e g[e] * Wq_fixed[e][f]
    {
        float acc = 0.0f;
        for (int e = 0; e < EE; ++e) acc += prevE[e] * Wq_fixed[e * EE + t];
        qfix[t] = acc;
    }

    // ------------- projections K1/V/K2 into LDS via fp32 WMMA -------------
    // 3 weights x 7 M-tiles(16) x 8 N-tiles(16) = 168 tiles, round-robin over 4 waves.
    {
        const int  lm = lane & 15;
        const bool hi = lane >= 16;
        for (int tile = wave; tile < 3 * 7 * 8; tile += 4) {
            const int wsel = tile / 56;
            const int rem  = tile % 56;
            const int m0   = (rem >> 3) * 16;
            const int n0   = (rem & 7) * 16;
            const float* W   = (wsel == 0) ? Wk1 : (wsel == 1) ? Wv : Wk2;
            float*       OUT = (wsel == 0) ? sK1 : (wsel == 1) ? sV : sK2;

            int m = m0 + lm; if (m > NN - 1) m = NN - 1;   // clamp pad rows
            const int n = n0 + lm;

            v8f c = {};
            for (int k0 = 0; k0 < EE; k0 += 4) {
                const int kb = k0 + (hi ? 2 : 0);
                // A 16x4 fp32: lanes 0-15 hold K=k0,k0+1; lanes 16-31 hold K=k0+2,k0+3
                v2f a, bb;
                a.x  = NEb[m * EE + kb];
                a.y  = NEb[m * EE + kb + 1];
                bb.x = W[kb * EE + n];
                bb.y = W[(kb + 1) * EE + n];
                c = __builtin_amdgcn_wmma_f32_16x16x4_f32(
                        /*neg_a=*/false, a, /*neg_b=*/false, bb,
                        /*c_mod=*/(short)0, c, /*reuse_a=*/false, /*reuse_b=*/false);
            }
            // C layout: lanes 0-15 -> rows m0+0..7, lanes 16-31 -> rows m0+8..15
            const int mbase = m0 + (hi ? 8 : 0);
            #pragma unroll
            for (int r = 0; r < 8; ++r) {
                const int mm = mbase + r;
                if (mm < NN) OUT[mm * LSTR + n] = c[r];
            }
        }
    }
#if HAVE_ASYNC_LDS
    __builtin_amdgcn_s_wait_asynccnt(0);
#endif
    __syncthreads();

    const float inv_sqrt_dh = 0.25f;
    const float inv_sqrt_e  = 0.08838834764831845f;   // 1/sqrt(128)

    // ---------------- decode loop: 202 serial steps ----------------
    for (int step = 0; step < TT; ++step) {
        // (a) previous node embedding (512B/step from L2)
        prevE[t] = NEb[sti[0] * EE + t];
        __syncthreads();

        // (b) q1[f] = qfix[f] + ctx @ Wq_step  (ctx = [prev_emb, D], 129 elems)
        {
            float acc = qfix[t];
            const float Dv = stf[0];
            for (int c = 0; c < EE; ++c) acc += prevE[c] * sWq[c * EE + t];
            acc += Dv * sWq[EE * EE + t];
            q1[t] = acc;
        }
        __syncthreads();

        // (c) masked attention scores: head h = t>>4, keys k = (t&15)+16j
        {
            const int h = t >> 4, sub = t & 15;
            const int prev = sti[0], vc = sti[1];
            const float Dv = stf[0];
            for (int k = sub; k < NN; k += 16) {
                bool m;
                if (k == 0) m = (prev == 0) && (vc < NCUST);
                else        m = (vis[k - 1] != 0) || (demS[k - 1] > Dv);
                float s;
                if (m) s = -1e9f;
                else {
                    s = 0.0f;
                    const float* kr = sK1 + k * LSTR + h * DHD;
                    const float* qr = q1 + h * DHD;
                    #pragma unroll
                    for (int d = 0; d < DHD; ++d) s += qr[d] * kr[d];
                    s *= inv_sqrt_dh;
                }
                attn[h * NN + k] = s;
            }
        }
        __syncthreads();

        // (d) per-head softmax in-place (16-lane groups, wave32 shfl width 16)
        {
            const int h = t >> 4, sub = t & 15;
            float mx = -INFINITY;
            for (int k = sub; k < NN; k += 16) mx = fmaxf(mx, attn[h * NN + k]);
            for (int off = 8; off; off >>= 1) mx = fmaxf(mx, __shfl_xor(mx, off, 16));
            float sm = 0.0f;
            for (int k = sub; k < NN; k += 16) {
                float e = expf(attn[h * NN + k] - mx);
                attn[h * NN + k] = e;
                sm += e;
            }
            for (int off = 8; off; off >>= 1) sm += __shfl_xor(sm, off, 16);
            const float inv = 1.0f / sm;
            for (int k = sub; k < NN; k += 16) attn[h * NN + k] *= inv;
        }
        __syncthreads();

        // (e) glimpse[h*16+d] = sum_k attn[h][k] * V[k][h*16+d]
        {
            const int h = t >> 4, d = t & 15;
            float g = 0.0f;
            for (int k = 0; k < NN; ++k) g += attn[h * NN + k] * sV[k * LSTR + h * DHD + d];
            glim[t] = g;
        }
        __syncthreads();

        // (f) q2 = glimpse @ Wout
        {
            float acc = 0.0f;
            for (int c = 0; c < EE; ++c) acc += glim[c] * sWo[c * EE + t];
            q2v[t] = acc;
        }
        __syncthreads();

        // (g) logits: thread t owns node k=t (t<101)
        float logit = -INFINITY;
        if (t < NN) {
            const int prev = sti[0], vc = sti[1];
            const float Dv = stf[0];
            bool m;
            if (t == 0) m = (prev == 0) && (vc < NCUST);
            else        m = (vis[t - 1] != 0) || (demS[t - 1] > Dv);
            if (m) logit = -1e9f;
            else {
                float acc = 0.0f;
                const float* kr = sK2 + t * LSTR;   // bank = (t + c) mod 64: conflict-free
                for (int c = 0; c < EE; ++c) acc += q2v[c] * kr[c];
                logit = 10.0f * tanhf(acc * inv_sqrt_e);
            }
        }

        // (h) argmax (first-max tie-break) + logsumexp across 128 threads
        float lv = logit; int li = t;
        for (int off = 16; off; off >>= 1) {
            float ov = __shfl_xor(lv, off, 32);
            int   oi = __shfl_xor(li, off, 32);
            if (ov > lv || (ov == lv && oi < li)) { lv = ov; li = oi; }
        }
        if (lane == 0) { redf[wave] = lv; redi[wave] = li; }
        __syncthreads();
        if (t == 0) {
            float bv = redf[0]; int bi = redi[0];
            for (int w = 1; w < 4; ++w) {
                float ov = redf[w]; int oi = redi[w];
                if (ov > bv || (ov == bv && oi < bi)) { bv = ov; bi = oi; }
            }
            redf[4] = bv; redi[4] = bi;
        }
        __syncthreads();
        const float bv = redf[4];
        float e = (logit == -INFINITY) ? 0.0f : expf(logit - bv);
        for (int off = 16; off; off >>= 1) e += __shfl_xor(e, off, 32);
        if (lane == 0) redf[wave] = e;
        __syncthreads();

        // (i) thread 0: state update, ll and tour-cost accumulation
        if (t == 0) {
            const float sume = redf[0] + redf[1] + redf[2] + redf[3];
            const int next = redi[4];
            stf[2] += -logf(sume);            // log_p[next] = bv - (bv + log(sume))

            const bool isDep = (next == 0);
            const int  cust  = isDep ? 0 : next - 1;
            const float Dn   = isDep ? 1.0f : (stf[0] - demS[cust]);
            if (!isDep && !vis[cust]) { vis[cust] = 1; sti[1] += 1; }

            float nx, ny;
            if (isDep) { nx = stf[5]; ny = stf[6]; }
            else {
                nx = customer_xy[((size_t)b * NCUST + cust) * 2 + 0];
                ny = customer_xy[((size_t)b * NCUST + cust) * 2 + 1];
            }
            const float dx = nx - stf[3], dy = ny - stf[4];
            stf[1] += sqrtf(dx * dx + dy * dy + 1e-10f);
            stf[3] = nx; stf[4] = ny; stf[0] = Dn; sti[0] = next;
        }
        __syncthreads();
    }

    // close the tour back to the depot, write outputs
    if (t == 0) {
        const float dx = stf[5] - stf[3], dy = stf[6] - stf[4];
        out[b]      = stf[1] + sqrtf(dx * dx + dy * dy + 1e-10f);
        out[NB + b] = stf[2];
    }
}

extern "C" void kernel_launch(void* const* d_in, const int* in_sizes, int n_in,
                              void* d_out, int out_size, void* d_ws, size_t ws_size,
                              hipStream_t stream) {
    (void)in_sizes; (void)n_in; (void)out_size; (void)d_ws; (void)ws_size;
    const float* depot_xy    = (const float*)d_in[0];
    const float* customer_xy = (const float*)d_in[1];
    const float* demand      = (const float*)d_in[2];
    const float* node_emb    = (const float*)d_in[3];
    const float* graph_emb   = (const float*)d_in[4];
    const float* Wk1         = (const float*)d_in[5];
    const float* Wv          = (const float*)d_in[6];
    const float* Wk2         = (const float*)d_in[7];
    const float* Wq_fixed    = (const float*)d_in[8];
    const float* Wout        = (const float*)d_in[9];
    const float* Wq_step     = (const float*)d_in[10];
    float* out = (float*)d_out;

    // dynamic LDS: weights (16512+16384) + 3*101*129 + 5*128 + 8*101 + 100 + 8
    // floats, + 110 ints + 7 floats  =  294,624 bytes (< 320 KB per WGP)
    const size_t shmem = (size_t)(NWQ + NWO
                                  + 3 * NN * LSTR + 5 * EE + HH * NN + NCUST + 8 + 7) * 4
                       + (size_t)(8 + NCUST + 2) * 4;

    vrp_decoder_kernel<<<dim3(NB), dim3(128), shmem, stream>>>(
        depot_xy, customer_xy, demand, node_emb, graph_emb,
        Wk1, Wv, Wk2, Wq_fixed, Wout, Wq_step, out);
}